// SpatialPrograms_50680614093476
// MI455X (gfx1250) — compile-verified
//
#include <hip/hip_runtime.h>
#include <hip/hip_bf16.h>

typedef __attribute__((ext_vector_type(2))) float v2f;
typedef __attribute__((ext_vector_type(8))) float v8f;

#define C_DIM 8
#define P_DIM 16
#define NG_DIM 5000
#define NS_DIM 5000
#define G_DIM 2000
#define S_DIM 2000
#define GT (G_DIM / 16)          // 125 gene tiles
#define ST (S_DIM / 16)          // 125 spot tiles
#define NTILES (GT * ST)         // 15625
#define WAVES_PER_BLOCK 8

__global__ __launch_bounds__(256) void spatial_programs_kernel(
    const float* __restrict__ log_rates,  // (C, NG, NS)
    const float* __restrict__ W,          // (C, P, NS)
    const float* __restrict__ V,          // (C, NG, P)
    const int*   __restrict__ spots,      // (S,)
    const int*   __restrict__ genes,      // (G,)
    float*       __restrict__ out)        // (G, S)
{
    const int lane = threadIdx.x & 31;
    const int wave = threadIdx.x >> 5;
    const int tile = blockIdx.x * WAVES_PER_BLOCK + wave;
    if (tile >= NTILES) return;           // wave-uniform: EXEC stays all-1s for WMMA

    const int gt = tile / ST;
    const int st = tile - gt * ST;
    const int g0 = gt * 16;
    const int s0 = st * 16;

    const int lo = lane & 15;             // tile column n (and A-row m)
    const int hi = lane >> 4;             // half-wave selector

    // Per-lane gather indices (wave reads 16 ints each -> fully L2/L0 resident)
    const int s_idx = spots[s0 + lo];     // spot for column n = lo
    const int gA    = genes[g0 + lo];     // gene row for A operand (M = lo)
    int grow[8];                          // gene rows matching C/D layout: M = 8*hi + v
#pragma unroll
    for (int v = 0; v < 8; ++v) grow[v] = genes[g0 + 8 * hi + v];

    v8f acc = {0.f, 0.f, 0.f, 0.f, 0.f, 0.f, 0.f, 0.f};

#pragma unroll 1
    for (int c = 0; c < C_DIM; ++c) {
        const float* Vrow = V + ((size_t)c * NG_DIM + gA) * P_DIM;   // 16 contiguous f32
        const float* Wc   = W + (size_t)c * P_DIM * NS_DIM;
        const float* LRc  = log_rates + (size_t)c * NG_DIM * NS_DIM;

        v8f low = {0.f, 0.f, 0.f, 0.f, 0.f, 0.f, 0.f, 0.f};

        // low(16x16) = Vg(16x16) x Wg(16x16) via 4 chained K=4 f32 WMMAs
#pragma unroll
        for (int k0 = 0; k0 < P_DIM; k0 += 4) {
            // A 16x4 f32 layout: VGPR0 = K {k0 | k0+2}, VGPR1 = K {k0+1 | k0+3}
            v2f a = *(const v2f*)(Vrow + k0 + 2 * hi);               // global_load_b64
            // B 4x16 f32 layout mirrors A striping across lanes
            v2f b;
            b.x = Wc[(size_t)(k0 + 2 * hi + 0) * NS_DIM + s_idx];
            b.y = Wc[(size_t)(k0 + 2 * hi + 1) * NS_DIM + s_idx];
            low = __builtin_amdgcn_wmma_f32_16x16x4_f32(
                /*neg_a=*/false, a, /*neg_b=*/false, b,
                /*c_mod=*/(short)0, low, /*reuse_a=*/false, /*reuse_b=*/false);
        }

        // Scattered lr gather lands directly in the WMMA C/D layout:
        // VGPR v <-> row M = 8*hi + v, lane%16 <-> column n
#pragma unroll
        for (int v = 0; v < 8; ++v) {
            const float lrv = LRc[(size_t)grow[v] * NS_DIM + s_idx];
            acc[v] += __expf(lrv + low[v]);
        }
    }

    // Store tile: lane owns column s0+lo, rows g0 + 8*hi + v (stride S)
    float* outp = out + (size_t)(g0 + 8 * hi) * S_DIM + s0 + lo;
#pragma unroll
    for (int v = 0; v < 8; ++v) outp[(size_t)v * S_DIM] = acc[v];
}

extern "C" void kernel_launch(void* const* d_in, const int* in_sizes, int n_in,
                              void* d_out, int out_size, void* d_ws, size_t ws_size,
                              hipStream_t stream) {
    const float* log_rates = (const float*)d_in[0];
    const float* W         = (const float*)d_in[1];
    const float* V         = (const float*)d_in[2];
    const int*   spots     = (const int*)d_in[3];
    const int*   genes     = (const int*)d_in[4];
    float*       out       = (float*)d_out;

    const int nblocks = (NTILES + WAVES_PER_BLOCK - 1) / WAVES_PER_BLOCK;  // 1954
    spatial_programs_kernel<<<nblocks, 256, 0, stream>>>(log_rates, W, V, spots, genes, out);
}